// IBPModel_8916352106568
// MI455X (gfx1250) — compile-verified
//
#include <hip/hip_runtime.h>

// MI455X / gfx1250 implementation of IBPModel.get_ub.
//
// One workgroup of 512 threads (16 wave32s) per batch row (B=65536, M=512).
//  - elementwise phase + two dot products (t, ret) via DS_SWIZZLE reductions
//  - stable descending sort of 512 ratio keys, packed u64
//    (ratio_bits<<32 | (511-idx)) == jnp's stable argsort(-value).
//    HYBRID bitonic: strides j<=16 entirely in registers via DS_SWIZZLE_B32
//    xor exchanges (1 ds op/dword, no address VALU, no barriers); the 10
//    steps with j>=32 use a DOUBLE-BUFFERED LDS exchange (1 barrier/step).
//  - prefix sum of sorted t_delta on the MATRIX pipe: inclusive scan ==
//    L(lower-tri ones) x S via chained V_WMMA_F32_16X16X4_F32 (exact f32),
//    2 waves x 4 wmmas cover all 512 elements; 32 group offsets stitched
//    with a wave-0 shuffle scan.
//  - fractional "percent" phase + final reduction
//
// Roofline: 256MB traffic (~11us @ 23.3TB/s); the sort is the VALU/latency
// bound, hence register-resident sort, swizzle exchanges, minimal barriers,
// and the WMMA-offloaded scan (co-executes with VALU).

#define EPS      1e-8f
#define FAKE_INF 10.0f
#define MDIM     512
#define BLOCK    512

typedef __attribute__((ext_vector_type(2))) float v2f;
typedef __attribute__((ext_vector_type(8))) float v8f;

__device__ __forceinline__
unsigned long long cmpex_sel(unsigned long long a, unsigned long long c,
                             bool keep_max)
{
    // keys are unique (index in low bits) -> strict compare is fine
    return (keep_max == (a > c)) ? a : c;
}

// Lane-xor exchange via DS_SWIZZLE_B32, group-of-32 mode:
// imm = {1'b0, xor_mask=J, or_mask=0, and_mask=0x1f}. One ds op per dword,
// no address computation (vs. ds_bpermute emitted for __shfl_xor).
template<int J>
__device__ __forceinline__ unsigned long long swz_xor_u64(unsigned long long x)
{
    constexpr int imm = 0x1f | (J << 10);
    union { unsigned long long u; int i[2]; } v;
    v.u = x;
    v.i[0] = __builtin_amdgcn_ds_swizzle(v.i[0], imm);
    v.i[1] = __builtin_amdgcn_ds_swizzle(v.i[1], imm);
    return v.u;
}

template<int J>
__device__ __forceinline__ float swz_xor_f32(float x)
{
    constexpr int imm = 0x1f | (J << 10);
    return __int_as_float(__builtin_amdgcn_ds_swizzle(__float_as_int(x), imm));
}

template<int J>
__device__ __forceinline__ void wave_reduce2(float& a, float& b)
{
    if constexpr (J >= 1) {
        a += swz_xor_f32<J>(a);
        b += swz_xor_f32<J>(b);
        wave_reduce2<(J >> 1)>(a, b);
    }
}

template<int J>
__device__ __forceinline__ void wave_reduce1(float& a)
{
    if constexpr (J >= 1) {
        a += swz_xor_f32<J>(a);
        wave_reduce1<(J >> 1)>(a);
    }
}

template<int KS, int J>
__device__ __forceinline__ void bitonic_shfl_tail(unsigned long long& key, int tid)
{
    if constexpr (J >= 1) {
        const unsigned long long c = swz_xor_u64<J>(key);
        const bool keep_max = (((tid & J) == 0) == ((tid & KS) == 0));
        key = cmpex_sel(key, c, keep_max);
        bitonic_shfl_tail<KS, (J >> 1)>(key, tid);
    }
}

template<int KS>
__device__ __forceinline__
void bitonic_stage(unsigned long long& key, int tid,
                   unsigned long long (&buf)[2][MDIM], int& ph)
{
    // Cross-wave strides: double-buffered LDS exchange, ONE barrier per step.
    // Safety: the buffer written here was last *read* two steps ago, and a
    // barrier (of the intervening step) separates those reads from this write.
    #pragma unroll
    for (int j = KS >> 1; j >= 32; j >>= 1) {
        buf[ph][tid] = key;
        __syncthreads();
        const unsigned long long c = buf[ph][tid ^ j];
        const bool keep_max = (((tid & j) == 0) == ((tid & KS) == 0));
        key = cmpex_sel(key, c, keep_max);
        ph ^= 1;
    }
    // In-wave strides: pure register/swizzle, no barriers.
    constexpr int JS = ((KS >> 1) < 16) ? (KS >> 1) : 16;
    bitonic_shfl_tail<KS, JS>(key, tid);
}

__global__ __launch_bounds__(BLOCK)
void ibp_get_ub_kernel(const float* __restrict__ K,
                       const float* __restrict__ K1,
                       const float* __restrict__ Bv,
                       const float* __restrict__ B1,
                       const float* __restrict__ WLB,
                       const float* __restrict__ WUB,
                       float* __restrict__ out)
{
    __shared__ unsigned long long s_keyb[2][MDIM];
    __shared__ float s_td[MDIM], s_t1d[MDIM], s_in[MDIM], s_scan[MDIM];
    __shared__ float s_gsum[32], s_goff[32];
    __shared__ float s_redt[BLOCK / 32], s_redr[BLOCK / 32], s_bcast[2];

    const int tid  = threadIdx.x;
    const int row  = blockIdx.x;
    const int lane = tid & 31;
    const int wv   = tid >> 5;

    // ---------------- elementwise phase (coalesced row load) ----------------
    const float k   = K [(size_t)row * MDIM + tid];
    const float k1  = K1[(size_t)row * MDIM + tid];
    const float wlb = WLB[tid];
    const float wub = WUB[tid];

    const bool  pick_lb = (k < 0.0f) || ((k == 0.0f) && (k1 < 0.0f));
    const float w_ret   = pick_lb ? wlb : wub;

    float pt = w_ret * k;    // -> t
    float pr = w_ret * k1;   // -> ret (pre-correction)

    const float span  = wub - wlb;
    const float ak    = fabsf(k);
    const float ak1   = fabsf(k1);
    const float td    = ak  * span;
    const float t1d   = ak1 * span;
    const float ratio = (k * k1 >= 0.0f) ? 0.0f : (ak1 / (EPS + ak));

    s_td[tid]  = td;
    s_t1d[tid] = t1d;

    // ratio >= 0 always -> IEEE bits order as unsigned ints.
    // Low bits (511 - idx): descending u64 sort == ratio desc, idx asc.
    unsigned long long key =
        ((unsigned long long)__float_as_uint(ratio) << 32)
        | (unsigned int)(MDIM - 1 - tid);

    // ---------------- reduce t and ret0 (swizzle + LDS) ---------------------
    wave_reduce2<16>(pt, pr);
    if (lane == 0) { s_redt[wv] = pt; s_redr[wv] = pr; }
    __syncthreads();   // also covers s_td/s_t1d stores
    if (tid == 0) {
        float st = 0.0f, sr = 0.0f;
        #pragma unroll
        for (int w = 0; w < BLOCK / 32; ++w) { st += s_redt[w]; sr += s_redr[w]; }
        s_bcast[0] = st + Bv[row];
        s_bcast[1] = sr + B1[row];
    }
    __syncthreads();
    const float t    = s_bcast[0];
    const float ret0 = s_bcast[1];

    // ---------------- hybrid bitonic sort (descending), keys in registers ---
    int ph = 0;
    bitonic_stage<  2>(key, tid, s_keyb, ph);
    bitonic_stage<  4>(key, tid, s_keyb, ph);
    bitonic_stage<  8>(key, tid, s_keyb, ph);
    bitonic_stage< 16>(key, tid, s_keyb, ph);
    bitonic_stage< 32>(key, tid, s_keyb, ph);
    bitonic_stage< 64>(key, tid, s_keyb, ph);
    bitonic_stage<128>(key, tid, s_keyb, ph);
    bitonic_stage<256>(key, tid, s_keyb, ph);
    bitonic_stage<512>(key, tid, s_keyb, ph);
    // Thread tid now holds the key of sorted position tid in a register.

    // ---------------- gather payloads in sorted order -----------------------
    const int   idx     = (MDIM - 1) - (int)(key & 0xFFFFFFFFu);
    const float ratio_s = __uint_as_float((unsigned int)(key >> 32));
    const float td_s    = s_td[idx];
    const float t1d_s   = s_t1d[idx];
    s_in[tid] = td_s;                 // scan input, 32 groups of 16
    __syncthreads();

    // ---------------- prefix sum on the matrix pipe --------------------------
    // Inclusive scan of each 16-element group g as one column of S:
    //   D = L x S,  L[m][kk] = (m >= kk),  S[kk][n] = s_in[G*16 + kk]
    // K=16 split into four V_WMMA_F32_16X16X4_F32 with accumulation.
    // f32 A 16x4 layout : lanes 0-15 M=0..15, vgpr j holds K = 2*(lane>>4)+j
    // f32 B 4x16 layout : lanes 0-15 N=0..15, vgpr j holds K = 2*(lane>>4)+j
    // f32 D 16x16 layout: lane gives N, vgpr v holds M = v + 8*(lane>>4)
    const int wvs = __builtin_amdgcn_readfirstlane(wv);  // scalar branch, EXEC all-1s
    if (wvs < 2) {
        const int mn = lane & 15;      // A row m == B column n for this lane
        const int kh = lane >> 4;      // lane-half selects K pair
        const int G  = wvs * 16 + mn;  // group (column) id, 0..31
        v8f acc = {0.f, 0.f, 0.f, 0.f, 0.f, 0.f, 0.f, 0.f};
        #pragma unroll
        for (int c = 0; c < 4; ++c) {
            const int k0 = 4 * c + 2 * kh;
            v2f a, b;
            a[0] = (mn >= k0)     ? 1.0f : 0.0f;   // lower-triangular ones
            a[1] = (mn >= k0 + 1) ? 1.0f : 0.0f;
            b[0] = s_in[G * 16 + k0];
            b[1] = s_in[G * 16 + k0 + 1];
            acc = __builtin_amdgcn_wmma_f32_16x16x4_f32(
                      false, a, false, b, (short)0, acc, false, false);
        }
        #pragma unroll
        for (int v = 0; v < 8; ++v) {
            const int m = v + 8 * kh;
            s_scan[G * 16 + m] = acc[v];           // inclusive intra-group scan
        }
        if (kh == 1) s_gsum[G] = acc[7];           // m = 15 -> group total
    }
    __syncthreads();
    if (wvs == 0) {                                // 32-wide exclusive scan, wave 0
        float x   = s_gsum[lane];
        float inc = x;
        #pragma unroll
        for (int off = 1; off < 32; off <<= 1) {
            const float y = __shfl_up(inc, off, 32);
            if (lane >= off) inc += y;
        }
        s_goff[lane] = inc - x;
    }
    __syncthreads();
    const float tds = s_scan[tid] + s_goff[tid >> 4];   // full cumsum

    // ---------------- percent phase + final reduction ------------------------
    float percent;
    const float delta = t - tds;
    if (delta < 0.0f) {
        float pcl = delta / (td_s + EPS);
        pcl = fmaxf(-1.0f, fminf(0.0f, pcl));
        percent = pcl + 1.0f;
    } else {
        percent = 1.0f;
    }
    if (!(ratio_s > 0.0f)) percent = 0.0f;

    float contrib = t1d_s * percent;
    wave_reduce1<16>(contrib);
    if (lane == 0) s_redt[wv] = contrib;
    __syncthreads();
    if (tid == 0) {
        float cs = 0.0f;
        #pragma unroll
        for (int w = 0; w < BLOCK / 32; ++w) cs += s_redt[w];
        out[row] = (t >= 0.0f) ? (ret0 + cs) : -FAKE_INF;
    }
}

extern "C" void kernel_launch(void* const* d_in, const int* in_sizes, int n_in,
                              void* d_out, int out_size, void* d_ws, size_t ws_size,
                              hipStream_t stream)
{
    (void)n_in; (void)d_ws; (void)ws_size; (void)out_size;
    const float* k   = (const float*)d_in[0];
    const float* k1  = (const float*)d_in[1];
    const float* b   = (const float*)d_in[2];
    const float* b1  = (const float*)d_in[3];
    const float* wlb = (const float*)d_in[4];
    const float* wub = (const float*)d_in[5];

    const int B = in_sizes[2];   // 65536 rows (b is [B])
    // Kernel is specialized for M == 512 (in_sizes[4]).

    ibp_get_ub_kernel<<<dim3(B), dim3(BLOCK), 0, stream>>>(
        k, k1, b, b1, wlb, wub, (float*)d_out);
}